// BackwardAveragePooling2D_13709535609394
// MI455X (gfx1250) — compile-verified
//
#include <hip/hip_runtime.h>

// BackwardAveragePooling2D (2x2, stride 2) == 2x nearest upsample with per-tap scale:
//   out[b, 2h+dh, 2w+dw, c] = x[b, h, w, c] * k[dh, dw]
// x: (32,112,112,128) f32, out: (32,224,224,128) f32.
// Pure HBM-bandwidth bound: ~205MB read + ~822MB write => ~44us floor @ 23.3 TB/s.
// Strategy (MI455X/gfx1250):
//   - wave32: 32 lanes x b128 = one pixel's 512B channel vector per wave
//   - async global->LDS loads (ASYNCcnt pipeline, no VGPRs held in flight)
//   - ds_load_b128 + VALU scale + 4x coalesced non-temporal b128 stores

typedef float v4f __attribute__((ext_vector_type(4)));

#define PPW 4              // pixels per wave (async pipeline depth)
#define WAVES_PER_BLOCK 8
#define BLOCK 256

#define C4 32              // 128 channels / 4 floats
#define W_IN 112
#define H_IN 112
#define PIX_PER_IMG (W_IN * H_IN)      // 12544
#define ROW_OUT (224 * 128)            // output row stride in floats

__global__ __launch_bounds__(BLOCK) void bwd_avgpool2x2_upsample(
    const float* __restrict__ x, const float* __restrict__ k,
    float* __restrict__ out)
{
    __shared__ v4f smem[WAVES_PER_BLOCK * PPW * C4];   // 16 KB

    const unsigned lane  = threadIdx.x & 31u;
    const unsigned wave  = threadIdx.x >> 5;
    const unsigned gwave = blockIdx.x * WAVES_PER_BLOCK + wave;
    const unsigned p0    = gwave * PPW;

    // kernel taps (uniform -> scalar loads); reference mapping: tap (dh,dw)
    // scales output (2h+dh, 2w+dw). All taps are 1/4 in the reference.
    const float k00 = k[0], k01 = k[1], k10 = k[2], k11 = k[3];

    // ---- Stage 1: issue PPW async global->LDS b128 loads (512B / pixel) ----
#pragma unroll
    for (int i = 0; i < PPW; ++i) {
        const unsigned p = p0 + i;
        const v4f* src = (const v4f*)x + (size_t)p * C4 + lane;
        // low 32 bits of the flat shared pointer == LDS byte offset;
        // ptrtoint also forces the compiler to treat smem as written by the asm.
        unsigned ldsOff = (unsigned)(size_t)&smem[(wave * PPW + i) * C4 + lane];
        asm volatile("global_load_async_to_lds_b128 %0, %1, off th:TH_LOAD_NT"
                     :: "v"(ldsOff), "v"(src)
                     : "memory");
    }

    // ---- Stage 2: drain in order (async loads complete in issue order) ----
#define STAGE(I, WREM)                                                         \
    do {                                                                       \
        asm volatile("s_wait_asynccnt " WREM ::: "memory");                    \
        const unsigned p = p0 + (I);                                           \
        v4f v = smem[(wave * PPW + (I)) * C4 + lane]; /* ds_load_b128 */       \
        const unsigned b = p / PIX_PER_IMG;                                    \
        const unsigned r = p % PIX_PER_IMG;                                    \
        const unsigned h = r / W_IN;                                           \
        const unsigned w = r % W_IN;                                           \
        float* base = out +                                                    \
            (((size_t)b * 224 + 2 * h) * 224 + 2 * w) * 128 + lane * 4;        \
        __builtin_nontemporal_store(v * k00, (v4f*)(base));                    \
        __builtin_nontemporal_store(v * k01, (v4f*)(base + 128));              \
        __builtin_nontemporal_store(v * k10, (v4f*)(base + ROW_OUT));          \
        __builtin_nontemporal_store(v * k11, (v4f*)(base + ROW_OUT + 128));    \
    } while (0)

    STAGE(0, "3");
    STAGE(1, "2");
    STAGE(2, "1");
    STAGE(3, "0");
#undef STAGE
}

extern "C" void kernel_launch(void* const* d_in, const int* in_sizes, int n_in,
                              void* d_out, int out_size, void* d_ws, size_t ws_size,
                              hipStream_t stream) {
    (void)in_sizes; (void)n_in; (void)out_size; (void)d_ws; (void)ws_size;
    const float* x = (const float*)d_in[0];
    const float* k = (const float*)d_in[1];
    float* out = (float*)d_out;

    // 32*112*112 = 401408 pixels; 4 per wave; 8 waves per block -> 12544 blocks
    const unsigned npix   = 32u * PIX_PER_IMG;
    const unsigned blocks = npix / (PPW * WAVES_PER_BLOCK);

    hipLaunchKernelGGL(bwd_avgpool2x2_upsample, dim3(blocks), dim3(BLOCK), 0,
                       stream, x, k, out);
}